// SparseDynamicConv3dKer_21517786153336
// MI455X (gfx1250) — compile-verified
//
#include <hip/hip_runtime.h>

// ---------------------------------------------------------------------------
// Sparse dynamic 3D conv: gather (masked) -> 48x48 fp32 GEMM per tap -> scatter-add
// MI455X / gfx1250, wave32, V_WMMA_F32_16X16X4_F32 for exact fp32 matrix math.
//
// v3 changes vs v2:
//  - no A-zeroing: WMMA rows are independent, so invalid rows just gather row 0
//    unconditionally and their (garbage) output row is skipped at scatter.
//    Removes the divergent branch + 24 v_cndmask per wave.
//  - weight staging uses shift/mask-only addressing (128-thread subgroup per
//    96-element interleaved pair-row), no integer-division magic.
//  - scatter validity compares hoisted (8 cmps reused across 3 tiles).
//  - float4 output clear.
// ---------------------------------------------------------------------------

typedef __attribute__((ext_vector_type(2))) float v2f;
typedef __attribute__((ext_vector_type(4))) float v4f;
typedef __attribute__((ext_vector_type(8))) float v8f;

#define KS_FULL   5
#define INC       64
#define OUTC      64
#define RUN_INC   48
#define RUN_OUTC  48
#define MPAIR     150000
#define K_RUN     27

#define BLOCK          256   // 8 waves (wave32)
#define ROWS_PER_WAVE  16
#define ROWS_PER_BLOCK 128   // 8 waves * 16 rows
#define KK_ELEMS       (RUN_INC * RUN_OUTC)   // 2304 = 24 pair-rows * 96

__global__ __launch_bounds__(BLOCK) void zero_f32v4(float* __restrict__ p, int n4) {
    int i = blockIdx.x * BLOCK + threadIdx.x;
    if (i < n4) ((v4f*)p)[i] = (v4f){0.0f, 0.0f, 0.0f, 0.0f};
}

__global__ __launch_bounds__(BLOCK) void spconv_wmma_f32(
    const float*         __restrict__ features,  // (N, 48)
    const float*         __restrict__ kernel,    // (125, 64, 64)
    const int*           __restrict__ in_idx,    // (27, M)
    const int*           __restrict__ out_idx,   // (27, M)
    const unsigned char* __restrict__ mask,      // (27, M) bool (1 byte)
    float*               __restrict__ out)       // (N, 48)
{
    // Pair-interleaved tap weights: kkp[pair*96 + col*2 + par] = w[2*pair+par][col]
    __shared__ float kkp[KK_ELEMS];              // 9216 B
    __shared__ int   s_inv [ROWS_PER_BLOCK];     // in row, or -1 (invalid/masked)
    __shared__ int   s_outv[ROWS_PER_BLOCK];     // out row, or -1

    const int k   = blockIdx.y;                  // tap 0..26
    const int m0  = blockIdx.x * ROWS_PER_BLOCK; // first pair of this block
    const int tid = threadIdx.x;

    // --- stage tap weights: slice [z+1][y+1][x+1][0:48][0:48] of (5,5,5,64,64).
    // 128-thread subgroups copy one 96-element pair-row each; addressing is
    // pure shift/mask (e = col*2 + par by construction).
    {
        const int kz = k / 9, ky = (k / 3) % 3, kx = k % 3;
        const int f  = ((kz + 1) * KS_FULL + (ky + 1)) * KS_FULL + (kx + 1);
        const float* ksrc = kernel + (size_t)f * (INC * OUTC);
        const int e  = tid & 127;                // position in pair-row
        const int p0 = tid >> 7;                 // 0 or 1
        if (e < 96) {
            const int col = e >> 1;
            const int par = e & 1;
            for (int j = 0; j < 12; ++j) {
                const int p = p0 + 2 * j;        // pair 0..23
                kkp[p * 96 + e] = ksrc[(2 * p + par) * OUTC + col];
            }
        }
    }

    // --- stage indices with validity folded in (-1 = skip row)
    if (tid < ROWS_PER_BLOCK) {
        const int m = m0 + tid;
        int ii = -1, oo = -1;
        if (m < MPAIR) {
            const size_t p = (size_t)k * MPAIR + m;
            if (mask[p]) { ii = in_idx[p]; oo = out_idx[p]; }
        }
        s_inv[tid] = ii; s_outv[tid] = oo;
    }
    __syncthreads();

    const int wave  = tid >> 5;     // 0..7
    const int lane  = tid & 31;
    const int lo    = lane & 15;    // row-within-half (A) / column (B,D)
    const int hi    = lane >> 4;    // lane half selects K+2 (A/B) or M+8 (D)
    const int rbase = wave * ROWS_PER_WAVE;

    // --- gather A fragments, unconditionally (invalid rows read row 0; their
    // garbage stays confined to their own D row, which scatter skips).
    // ISA A layout (16x4 f32): lanes 0-15 = rows M; VGPR0 = K / K+2 by lane
    // half, VGPR1 = K+1 / K+3.  a[t] = feat[row][4t + 2*hi + {0,1}].
    v2f a[12];
    {
        const int ir = s_inv[rbase + lo];
        const int rr = ir >= 0 ? ir : 0;
        const float* src = features + (size_t)rr * RUN_INC + 2 * hi;
#pragma unroll
        for (int t = 0; t < 12; ++t)
            a[t] = *(const v2f*)(src + 4 * t);   // global_load_b64, imm offset
    }

    // --- precompute scatter addresses + validity (rows v + 8*hi, column lo);
    // per-tile column offset 16*n floats folds into the atomic's imm offset.
    bool   ok  [8];
    float* addr[8];
#pragma unroll
    for (int v = 0; v < 8; ++v) {
        const int orow = s_outv[rbase + v + 8 * hi];
        ok[v]   = (orow >= 0);
        addr[v] = out + (size_t)(orow >= 0 ? orow : 0) * RUN_OUTC + lo;
    }

    // --- 3 output column tiles of 16; K=48 via 12 chained WMMAs each
#pragma unroll
    for (int n = 0; n < 3; ++n) {
        v8f c = {};
#pragma unroll
        for (int t = 0; t < 12; ++t) {
            // B layout mirrors A: pair-row 2t+hi supplies K rows 4t+2hi, +1;
            // columns striped across lanes 0-15.  One ds_load_b64 per step,
            // conflict-free across all 64 banks.
            const v2f b = *(const v2f*)&kkp[(2 * t + hi) * 96 + (16 * n + lo) * 2];
            c = __builtin_amdgcn_wmma_f32_16x16x4_f32(
                    /*neg_a=*/false, a[t], /*neg_b=*/false, b,
                    /*c_mod=*/(short)0, c, /*reuse_a=*/false, /*reuse_b=*/false);
        }
        // --- scatter-add: D layout 16x16 f32 -> VGPR v = row v / v+8 by lane
        // half, column = lane lo.
#pragma unroll
        for (int v = 0; v < 8; ++v) {
            if (ok[v]) {
                atomicAdd(addr[v] + 16 * n, c[v]);   // global_atomic_add_f32
            }
        }
    }
}

extern "C" void kernel_launch(void* const* d_in, const int* in_sizes, int n_in,
                              void* d_out, int out_size, void* d_ws, size_t ws_size,
                              hipStream_t stream) {
    const float*         features = (const float*)d_in[0];
    const float*         kernelw  = (const float*)d_in[1];
    const int*           in_idx   = (const int*)d_in[2];
    const int*           out_idx  = (const int*)d_in[3];
    const unsigned char* mask     = (const unsigned char*)d_in[4]; // jax bool = 1 byte
    float*               out      = (float*)d_out;

    // d_out is poisoned by the harness -> zero it first (same stream = ordered).
    // out_size = 200000*48 is divisible by 4 -> float4 clear.
    {
        const int n4      = out_size >> 2;
        const int nblocks = (n4 + BLOCK - 1) / BLOCK;
        zero_f32v4<<<nblocks, BLOCK, 0, stream>>>(out, n4);
    }

    dim3 grid((MPAIR + ROWS_PER_BLOCK - 1) / ROWS_PER_BLOCK, K_RUN);
    spconv_wmma_f32<<<grid, BLOCK, 0, stream>>>(features, kernelw, in_idx, out_idx,
                                                mask, out);
}